// LowRankGCNConv_14697378087196
// MI455X (gfx1250) — compile-verified
//
#include <hip/hip_runtime.h>

typedef __attribute__((ext_vector_type(2))) float v2f;
typedef __attribute__((ext_vector_type(8))) float v8f;

#define D128 128

// ---------------------------------------------------------------------------
// Kernel 1: init accumulators. deg starts at 1.0 (self-loop), alpha/beta at 0.
// ---------------------------------------------------------------------------
__global__ void k_init(float* __restrict__ deg, float* __restrict__ aAcc,
                       float* __restrict__ bAcc, int n) {
    int i = blockIdx.x * blockDim.x + threadIdx.x;
    if (i < n) {
        deg[i]  = 1.0f;   // self-loop contributes 1 to in-degree of col
        aAcc[i] = 0.0f;
        bAcc[i] = 0.0f;
    }
}

// ---------------------------------------------------------------------------
// Kernel 2: s[i] = dot(x[i,:], a1).  One wave32 per node; lane l owns the
// float4 at dims [4l..4l+3] -> a single coalesced 512B row read per wave.
// ---------------------------------------------------------------------------
__global__ void k_dot(const float* __restrict__ x, const float* __restrict__ a1,
                      float* __restrict__ s, int n) {
    int lane = threadIdx.x & 31;
    int wave = (blockIdx.x * (blockDim.x >> 5)) + (threadIdx.x >> 5);
    if (wave >= n) return;

    const float4 xr = ((const float4*)x)[(size_t)wave * (D128 / 4) + lane];
    const float4 ar = ((const float4*)a1)[lane];
    float v = xr.x * ar.x + xr.y * ar.y + xr.z * ar.z + xr.w * ar.w;

    // wave32 butterfly reduction
    #pragma unroll
    for (int off = 16; off > 0; off >>= 1)
        v += __shfl_xor(v, off, 32);

    if (lane == 0) s[wave] = v;
}

// ---------------------------------------------------------------------------
// Kernel 3: degree accumulation over edges (L2 scalar atomics; the 200KB
// degree array is fully resident in the 192MB L2).
// ---------------------------------------------------------------------------
__global__ void k_deg(const int* __restrict__ col, float* __restrict__ deg, int e) {
    int i = blockIdx.x * blockDim.x + threadIdx.x;
    int stride = gridDim.x * blockDim.x;
    for (; i < e; i += stride)
        atomicAdd(&deg[col[i]], 1.0f);
}

// ---------------------------------------------------------------------------
// Kernel 4: dinv = rsqrt(deg) (+1 Newton step for accuracy), w = dinv*s.
// dinv overwrites deg in place.
// ---------------------------------------------------------------------------
__global__ void k_dinv(float* __restrict__ deg_dinv, const float* __restrict__ s,
                       float* __restrict__ w, int n) {
    int i = blockIdx.x * blockDim.x + threadIdx.x;
    if (i < n) {
        float d = deg_dinv[i];                 // >= 1 always (self-loop)
        float r = __frsqrt_rn(d);
        r = r * (1.5f - 0.5f * d * r * r);     // Newton refine
        deg_dinv[i] = r;
        w[i] = r * s[i];
    }
}

// ---------------------------------------------------------------------------
// Kernel 5: per-edge scalar scatter:
//   aAcc[row] += w[col]   (= dinv[col]*s[col])
//   bAcc[row] += dinv[col]
// Gathers of w/dinv are random but L2-resident.
// ---------------------------------------------------------------------------
__global__ void k_scatter(const int* __restrict__ row, const int* __restrict__ col,
                          const float* __restrict__ dinv, const float* __restrict__ w,
                          float* __restrict__ aAcc, float* __restrict__ bAcc, int e) {
    int i = blockIdx.x * blockDim.x + threadIdx.x;
    int stride = gridDim.x * blockDim.x;
    for (; i < e; i += stride) {
        int r = row[i];
        int c = col[i];
        atomicAdd(&aAcc[r], w[c]);
        atomicAdd(&bAcc[r], dinv[c]);
    }
}

// ---------------------------------------------------------------------------
// Kernel 6: out[r,d] = alpha[r]*a2[d] + beta[r]*bias[d] via WMMA f32 16x16x4.
// Exactly a 16x2 @ 2x16 matmul per tile: A = [alpha beta 0 0], B rows =
// [a2 ; bias ; 0 ; 0]. One wave produces a 16-node x 128-dim slab with 8
// v_wmma_f32_16x16x4_f32 ops.
//
// A layout (32-bit A 16x4): lanes 0-15 = rows M=0..15, VGPR0=K0, VGPR1=K1;
//                           lanes 16-31 hold K2/K3 (we feed zeros).
// B layout (mirrored):      lanes 0-15 = cols N=0..15, VGPR0=K0 row, VGPR1=K1;
//                           lanes 16-31 = K2/K3 rows (zeros).
// C/D layout: VGPR v holds row (v + 8*half), col = lane&15.
//
// Store guard is wave-uniform (full slab vs ragged tail) so the hot path is a
// straight-line clause of 8 coalesced b32 stores per tile with EXEC all-ones.
// ---------------------------------------------------------------------------
__global__ void k_out_wmma(const float* __restrict__ dinv, const float* __restrict__ s_w,
                           const float* __restrict__ aAcc, const float* __restrict__ bAcc,
                           const float* __restrict__ a2, const float* __restrict__ bias,
                           float* __restrict__ out, int n) {
    const int lane = threadIdx.x & 31;
    const int wave = (blockIdx.x * (blockDim.x >> 5)) + (threadIdx.x >> 5);
    const int base = wave * 16;
    if (base >= n) return;   // uniform per wave -> EXEC stays all-ones below

    // ---- A fragment: per-node alpha/beta in lanes 0-15, zeros for K2/K3 ----
    float alpha = 0.0f, beta = 0.0f;
    if (lane < 16) {
        int node = base + lane;
        if (node < n) {
            float di = dinv[node];
            alpha = di * (aAcc[node] + s_w[node]);   // s_w = w = dinv*s
            beta  = di * (bAcc[node] + di);
        }
    }
    v2f afrag;
    afrag.x = alpha;   // K=0 (lanes 0-15) / K=2 = 0 (lanes 16-31)
    afrag.y = beta;    // K=1 (lanes 0-15) / K=3 = 0 (lanes 16-31)

    const int half = lane >> 4;      // 0 or 1
    const int ln   = lane & 15;
    const bool full_slab = (base + 16 <= n);   // wave-uniform
    // per-lane base address for C/D stores: row (base + 8*half), col d0+ln
    float* outp = out + (size_t)(base + 8 * half) * D128 + ln;

    #pragma unroll
    for (int t = 0; t < 8; ++t) {
        const int d0 = t * 16;

        // ---- B fragment: a2/bias rows in lanes 0-15, zeros for K2/K3 ----
        v2f bfrag;
        float b0 = 0.0f, b1 = 0.0f;
        if (lane < 16) {
            b0 = a2[d0 + lane];
            b1 = bias[d0 + lane];
        }
        bfrag.x = b0;
        bfrag.y = b1;

        v8f c = {};
        c = __builtin_amdgcn_wmma_f32_16x16x4_f32(
                /*neg_a=*/false, afrag, /*neg_b=*/false, bfrag,
                /*c_mod=*/(short)0, c, /*reuse_a=*/false, /*reuse_b=*/false);

        if (full_slab) {
            // hot path: straight-line coalesced stores, no per-row branches
            #pragma unroll
            for (int v = 0; v < 8; ++v)
                outp[(size_t)v * D128 + d0] = c[v];
        } else {
            // ragged tail (never taken when n % 16 == 0)
            #pragma unroll
            for (int v = 0; v < 8; ++v) {
                int node_row = base + v + 8 * half;
                if (node_row < n)
                    outp[(size_t)v * D128 + d0] = c[v];
            }
        }
    }
}

// ---------------------------------------------------------------------------
// Host-side orchestration
// ---------------------------------------------------------------------------
extern "C" void kernel_launch(void* const* d_in, const int* in_sizes, int n_in,
                              void* d_out, int out_size, void* d_ws, size_t ws_size,
                              hipStream_t stream) {
    (void)n_in; (void)out_size; (void)ws_size;

    const float* x    = (const float*)d_in[0];
    const int*   ei   = (const int*)d_in[1];   // [2, E]: row-major
    const float* a1   = (const float*)d_in[2];
    const float* a2   = (const float*)d_in[3];
    const float* bias = (const float*)d_in[4];
    float*       out  = (float*)d_out;

    const int N = in_sizes[0] / D128;
    const int E = in_sizes[1] / 2;
    const int* row = ei;
    const int* col = ei + E;

    // workspace: 5 float arrays of length N  (5*N*4 = ~1 MB)
    float* s    = (float*)d_ws;
    float* deg  = s    + N;   // becomes dinv in place after k_dinv
    float* w    = deg  + N;
    float* aAcc = w    + N;
    float* bAcc = aAcc + N;

    const int TB = 256;

    // 1) init
    k_init<<<(N + TB - 1) / TB, TB, 0, stream>>>(deg, aAcc, bAcc, N);

    // 2) s = x @ a1  (one wave per node, 8 waves per block)
    {
        int waves_per_block = TB / 32;
        int blocks = (N + waves_per_block - 1) / waves_per_block;
        k_dot<<<blocks, TB, 0, stream>>>(x, a1, s, N);
    }

    // 3) degree atomics
    {
        int blocks = (E + TB - 1) / TB;
        if (blocks > 4096) blocks = 4096;
        k_deg<<<blocks, TB, 0, stream>>>(col, deg, E);
    }

    // 4) dinv + w
    k_dinv<<<(N + TB - 1) / TB, TB, 0, stream>>>(deg, s, w, N);

    // 5) edge scatter for alpha/beta
    {
        int blocks = (E + TB - 1) / TB;
        if (blocks > 4096) blocks = 4096;
        k_scatter<<<blocks, TB, 0, stream>>>(row, col, deg /*=dinv*/, w, aAcc, bAcc, E);
    }

    // 6) WMMA rank-2 reconstruction: out = [alpha|beta] @ [a2 ; bias]
    {
        int waves = (N + 15) / 16;
        int waves_per_block = TB / 32;
        int blocks = (waves + waves_per_block - 1) / waves_per_block;
        k_out_wmma<<<blocks, TB, 0, stream>>>(deg /*=dinv*/, w, aAcc, bAcc,
                                              a2, bias, out, N);
    }
}